// CausalConv1d_85504208929343
// MI455X (gfx1250) — compile-verified
//
#include <hip/hip_runtime.h>
#include <stdint.h>

// Problem constants (from the reference): x (4, 4096, 2048) f32, w (2048,1,4) f32.
#define B_   4
#define S_   4096
#define D_   2048

// Tiling: one block = 256 channels x 64 s-rows. LDS tile = 64*256*4 = 64 KB.
#define S_TILE        64
#define D_TILE        256
#define WAVES_PER_BLK 8
#define ROWS_PER_WAVE (S_TILE / WAVES_PER_BLK)   // 8 rows per wave TDM slice

typedef __attribute__((ext_vector_type(4))) unsigned int u32x4;
typedef __attribute__((ext_vector_type(8))) int          i32x8;
typedef __attribute__((ext_vector_type(4))) int          i32x4;

__global__ __launch_bounds__(256) void causal_dwconv_silu_kernel(
    const float* __restrict__ x, const float* __restrict__ w,
    float* __restrict__ out)
{
    __shared__ float tile[S_TILE * D_TILE];   // 64 KB

    const int tid = threadIdx.x;              // channel within d-tile
    const int s0  = blockIdx.x * S_TILE;      // tile start along S
    const int d0  = blockIdx.y * D_TILE;      // tile start along D
    const int b   = blockIdx.z;
    const int d   = d0 + tid;                 // this thread's channel

    // ---------------- TDM: each wave DMAs its 8-row slice into LDS ----------
    // Tile rows are 1 KB contiguous (256 ch * 4B) with 8 KB stride (D=2048 f32).
    const int wave = __builtin_amdgcn_readfirstlane((int)(threadIdx.x >> 5));
    const int srow = s0 + wave * ROWS_PER_WAVE;

    unsigned long long ga = (unsigned long long)(uintptr_t)
        (x + ((size_t)b * S_ + (size_t)srow) * D_ + d0);
    unsigned lds_base = (unsigned)(uintptr_t)(void*)&tile[0];
    unsigned lds_addr = lds_base + (unsigned)(wave * ROWS_PER_WAVE * D_TILE * 4);

    // D# group 0 (ISA 8.3): count=1 | lds_addr | global_addr(57b) | type=2
    u32x4 g0;
    g0.x = 1u;                                     // count=1, user descriptor
    g0.y = lds_addr;                               // bits [63:32] lds_addr
    g0.z = (unsigned)(ga & 0xFFFFFFFFu);           // bits [95:64] ga lo
    g0.w = ((unsigned)(ga >> 32) & 0x01FFFFFFu)    // bits [120:96] ga hi
         | (2u << 30);                             // bits [127:126] type=2

    // D# group 1 (ISA 8.4): data_size=4B, tensor 2048 x (B*S), tile 256 x 8,
    // dim0 stride = 2048 elements. No multicast / pad / iterate / barrier.
    i32x8 g1;
    g1[0] = (int)(2u << 16);                 // data_size = 2 (4 bytes)
    g1[1] = (int)((unsigned)D_ << 16);       // tensor_dim0 lo16 at [63:48]
    g1[2] = (int)((unsigned)(B_ * S_) << 16);// tensor_dim1 lo16; dim0 hi16 = 0
    g1[3] = (int)((unsigned)D_TILE << 16);   // tile_dim0 = 256; dim1 hi16 = 0
    g1[4] = ROWS_PER_WAVE;                   // tile_dim1 = 8; tile_dim2 = 0
    g1[5] = D_;                              // tensor_dim0_stride lo32 = 2048
    g1[6] = 0;                               // stride0 hi16 = 0; stride1 lo16 = 0
    g1[7] = 0;

    i32x4 gz = {0, 0, 0, 0};                 // groups 2/3 unused (2D tensor)

#if defined(__clang_major__) && (__clang_major__ >= 23)
    i32x8 gz8 = {0, 0, 0, 0, 0, 0, 0, 0};
    __builtin_amdgcn_tensor_load_to_lds(g0, g1, gz, gz, gz8, 0);
#else
    __builtin_amdgcn_tensor_load_to_lds(g0, g1, gz, gz, 0);
#endif

    // -------- overlap with the DMA: weights + causal halo (3 rows) ----------
    // w layout (D, 1, 4): 16B per channel, 16B aligned -> b128 load.
    float4 wv = *(const float4*)(w + (size_t)d * 4);

    const float* xb = x + (size_t)b * S_ * D_ + d;
    float h0 = 0.0f, h1 = 0.0f, h2 = 0.0f;    // x[s0-3], x[s0-2], x[s0-1]
    if (s0 > 0) {                              // s0 is a multiple of 64
        h0 = xb[(size_t)(s0 - 3) * D_];
        h1 = xb[(size_t)(s0 - 2) * D_];
        h2 = xb[(size_t)(s0 - 1) * D_];
    }

    __builtin_amdgcn_s_wait_tensorcnt(0);     // my slice landed in LDS
    __syncthreads();                          // all slices landed

    // -------- sliding-window FIR + SiLU, one channel per lane ---------------
    float* orow = out + ((size_t)b * S_ + s0) * D_ + d;
    const float* trow = &tile[tid];

    #pragma unroll 8
    for (int r = 0; r < S_TILE; ++r) {
        float xc  = trow[(size_t)r * D_TILE];                 // ds_load_b32
        float acc = fmaf(wv.x, h0,
                    fmaf(wv.y, h1,
                    fmaf(wv.z, h2, wv.w * xc)));
        // silu(acc) = acc * sigmoid(acc) = acc / (1 + exp(-acc))
        float sg = __builtin_amdgcn_rcpf(1.0f + __expf(-acc));
        __builtin_nontemporal_store(acc * sg, orow + (size_t)r * D_);
        h0 = h1; h1 = h2; h2 = xc;
    }
}

extern "C" void kernel_launch(void* const* d_in, const int* in_sizes, int n_in,
                              void* d_out, int out_size, void* d_ws, size_t ws_size,
                              hipStream_t stream)
{
    (void)in_sizes; (void)n_in; (void)out_size; (void)d_ws; (void)ws_size;
    const float* x = (const float*)d_in[0];
    const float* w = (const float*)d_in[1];
    float* out = (float*)d_out;

    dim3 grid(S_ / S_TILE, D_ / D_TILE, B_);  // 64 x 8 x 4 = 2048 blocks
    dim3 block(256);                          // 8 wave32 per block
    causal_dwconv_silu_kernel<<<grid, block, 0, stream>>>(x, w, out);
}